// PersistentHomologyLayer_56573309223825
// MI455X (gfx1250) — compile-verified
//
#include <hip/hip_runtime.h>
#include <hip/hip_bf16.h>
#include <math.h>

#define NPTS 8192
#define DIMS 64
#define BIGF 1e30f

typedef __attribute__((ext_vector_type(2))) float v2f;
typedef __attribute__((ext_vector_type(8))) float v8f;

// ---------------------------------------------------------------------------
// Kernel 1: per-row squared norms  sq[i] = sum_k X[i][k]^2
// ---------------------------------------------------------------------------
__global__ __launch_bounds__(256) void rowsq_kernel(const float* __restrict__ X,
                                                    float* __restrict__ sq) {
    int i = blockIdx.x * blockDim.x + threadIdx.x;
    if (i < NPTS) {
        const float* r = X + (size_t)i * DIMS;
        float s = 0.0f;
#pragma unroll
        for (int k = 0; k < DIMS; ++k) s = fmaf(r[k], r[k], s);
        sq[i] = s;
    }
}

// ---------------------------------------------------------------------------
// Kernel 2: distance matrix via fp32 WMMA (V_WMMA_F32_16X16X4_F32).
// One wave computes a 1x4 strip of 16x16 tiles: each k-step loads ONE A
// fragment and four B fragments feeding FOUR WMMAs (A reuse cuts vmem issue
// from 2 loads/WMMA to 1.25 loads/WMMA).
//
// A-frag (16x4 f32): lanes 0-15 hold M=lane, K={4k,4k+1}; lanes 16-31 hold
// M=lane-16, K={4k+2,4k+3}. B-frag (4x16) mirrors it with the column index.
// C/D (16x16 f32, 8 VGPRs): c[r] is M = r + 8*(lane>=16), N = lane&15.
// ---------------------------------------------------------------------------
__global__ __launch_bounds__(256) void dist_wmma_kernel(const float* __restrict__ X,
                                                        const float* __restrict__ sq,
                                                        float* __restrict__ D) {
    const int lane = threadIdx.x;                   // 0..31 (wave32)
    const int wib  = threadIdx.y;                   // 0..7 waves per block
    const int ti   = blockIdx.x;                    // row tile 0..511
    const int tjs  = (blockIdx.y * 8 + wib) * 4;    // col tile strip base
    const int l15  = lane & 15;
    const int hi   = lane >> 4;                     // 0: K pair {0,1}; 1: {2,3}

    const float* Arow = X + ((size_t)(ti * 16 + l15)) * DIMS + 2 * hi;
    const float* Brow = X + ((size_t)(tjs * 16 + l15)) * DIMS + 2 * hi;

    v8f cacc[4] = {};
#pragma unroll
    for (int k = 0; k < DIMS / 4; ++k) {
        v2f a = *(const v2f*)(Arow + 4 * k);
#pragma unroll
        for (int s = 0; s < 4; ++s) {
            v2f b = *(const v2f*)(Brow + (size_t)s * 16 * DIMS + 4 * k);
            cacc[s] = __builtin_amdgcn_wmma_f32_16x16x4_f32(
                          /*neg_a=*/false, a, /*neg_b=*/false, b,
                          /*c_mod=*/(short)0, cacc[s],
                          /*reuse_a=*/false, /*reuse_b=*/false);
        }
    }

    // epilogue: d = sqrt(max(sq_i + sq_j - 2*dot, 0))
    float sqm[8];
#pragma unroll
    for (int r = 0; r < 8; ++r) sqm[r] = sq[ti * 16 + r + 8 * hi];

#pragma unroll
    for (int s = 0; s < 4; ++s) {
        const int   n   = (tjs + s) * 16 + l15;
        const float sqn = sq[n];
        float* ob = D + (size_t)(ti * 16) * NPTS + n;
#pragma unroll
        for (int r = 0; r < 8; ++r) {
            int   m  = r + 8 * hi;
            float d2 = sqm[r] + sqn - 2.0f * cacc[s][r];
            ob[(size_t)m * NPTS] = sqrtf(fmaxf(d2, 0.0f));
        }
    }
}

// ---------------------------------------------------------------------------
// Kernel 3: Prim's MST scan, single workgroup (32 waves), LDS-resident state.
// Per round: per-thread argmin over 8 strided entries -> wave32 butterfly
// argmin via __shfl_xor (ds-permute datapath, no barriers) -> 32 partials in
// LDS -> wave 0 butterfly -> broadcast. 3 __syncthreads per round.
// Tie-break: lowest index, matching jnp.argmin.
// ---------------------------------------------------------------------------
__global__ __launch_bounds__(1024) void prim_kernel(const float* __restrict__ D,
                                                    float* __restrict__ out) {
    __shared__ float         mind[NPTS];
    __shared__ unsigned char intree[NPTS];
    __shared__ float         wval[32];
    __shared__ int           widx[32];
    __shared__ int           jsel;

    const int tid = threadIdx.x;

    // init: mind = dist row 0, tree = {0}
#pragma unroll
    for (int r = 0; r < NPTS / 1024; ++r) {
        int i = tid + r * 1024;
        mind[i]   = D[i];
        intree[i] = 0;
    }
    __syncthreads();
    if (tid == 0) intree[0] = 1;
    __syncthreads();

    for (int t = 0; t < NPTS - 1; ++t) {
        // --- per-thread argmin over strided entries (strict < keeps low idx)
        float bv = BIGF;
        int   bi = 0;
#pragma unroll
        for (int r = 0; r < NPTS / 1024; ++r) {
            int   i = tid + r * 1024;
            float v = intree[i] ? BIGF : mind[i];
            if (v < bv) { bv = v; bi = i; }
        }

        // --- wave32 butterfly argmin (associative op -> exact global argmin)
#pragma unroll
        for (int off = 16; off > 0; off >>= 1) {
            float ov = __shfl_xor(bv, off, 32);
            int   oi = __shfl_xor(bi, off, 32);
            if (ov < bv || (ov == bv && oi < bi)) { bv = ov; bi = oi; }
        }
        if ((tid & 31) == 0) {
            wval[tid >> 5] = bv;
            widx[tid >> 5] = bi;
        }
        __syncthreads();

        // --- wave 0 reduces the 32 per-wave partials
        if (tid < 32) {
            bv = wval[tid];
            bi = widx[tid];
#pragma unroll
            for (int off = 16; off > 0; off >>= 1) {
                float ov = __shfl_xor(bv, off, 32);
                int   oi = __shfl_xor(bi, off, 32);
                if (ov < bv || (ov == bv && oi < bi)) { bv = ov; bi = oi; }
            }
            if (tid == 0) {
                jsel = bi;
                intree[bi] = 1;
                out[2 * t + 0] = 0.0f;  // birth
                out[2 * t + 1] = bv;    // death = MST edge weight
            }
        }
        __syncthreads();

        // --- relax: mind = min(mind, dist[j])  (32KB coalesced row stream)
        const int    j   = jsel;
        const float* row = D + (size_t)j * NPTS;
#pragma unroll
        for (int r = 0; r < NPTS / 1024; ++r) {
            int i   = tid + r * 1024;
            mind[i] = fminf(mind[i], row[i]);
        }
        __syncthreads();
    }
}

// ---------------------------------------------------------------------------
// launcher
// ---------------------------------------------------------------------------
extern "C" void kernel_launch(void* const* d_in, const int* in_sizes, int n_in,
                              void* d_out, int out_size, void* d_ws, size_t ws_size,
                              hipStream_t stream) {
    (void)in_sizes; (void)n_in; (void)out_size; (void)ws_size;

    const float* x   = (const float*)d_in[0];           // [8192, 64] fp32
    float*       out = (float*)d_out;                   // [8191, 2]  fp32
    float*       sq  = (float*)d_ws;                    // 8192 floats
    float*       Dm  = (float*)((char*)d_ws + NPTS * sizeof(float)); // 8192^2 floats

    rowsq_kernel<<<NPTS / 256, 256, 0, stream>>>(x, sq);

    // 512 row tiles x 128 column strips (each strip = 4 tiles), 8 waves/block
    dist_wmma_kernel<<<dim3(NPTS / 16, NPTS / 16 / 4 / 8), dim3(32, 8), 0, stream>>>(x, sq, Dm);

    prim_kernel<<<1, 1024, 0, stream>>>(Dm, out);
}